// GraphSAGEFeat_87282325390023
// MI455X (gfx1250) — compile-verified
//
#include <hip/hip_runtime.h>

typedef __attribute__((ext_vector_type(2))) float v2f;
typedef __attribute__((ext_vector_type(8))) float v8f;

#define N_NODES 100000
#define N_EDGES 1600000
#define IN_F 128
#define HID_F 128
#define OUT_F 64

// ---------------------------------------------------------------------------
// Zero-fill (float4 granularity)
// ---------------------------------------------------------------------------
__global__ void zero_kernel(float4* __restrict__ p, int n4) {
    int i = blockIdx.x * blockDim.x + threadIdx.x;
    if (i < n4) p[i] = make_float4(0.f, 0.f, 0.f, 0.f);
}

// ---------------------------------------------------------------------------
// Edge scatter: one wave32 per edge. 32 lanes x float4 = 128 features.
// msg[dst] += h[src]; optionally deg[dst] += 1 (float).
// msg buffer (51.2MB) is L2-resident on MI455X (192MB L2) -> atomics stay in L2.
// ---------------------------------------------------------------------------
__global__ __launch_bounds__(256)
void scatter_kernel(const float* __restrict__ h,
                    const int* __restrict__ src,
                    const int* __restrict__ dst,
                    float* __restrict__ msg,
                    float* __restrict__ deg,
                    int n_edges, int count_deg) {
    const int lane = threadIdx.x & 31;
    const int e = blockIdx.x * (blockDim.x >> 5) + (threadIdx.x >> 5);
    if (e >= n_edges) return;
    const int s = src[e];
    const int d = dst[e];
    const float4 v = ((const float4*)(h + (size_t)s * 128))[lane];  // coalesced 512B/wave
    float* mrow = msg + (size_t)d * 128 + lane * 4;
    __hip_atomic_fetch_add(mrow + 0, v.x, __ATOMIC_RELAXED, __HIP_MEMORY_SCOPE_AGENT);
    __hip_atomic_fetch_add(mrow + 1, v.y, __ATOMIC_RELAXED, __HIP_MEMORY_SCOPE_AGENT);
    __hip_atomic_fetch_add(mrow + 2, v.z, __ATOMIC_RELAXED, __HIP_MEMORY_SCOPE_AGENT);
    __hip_atomic_fetch_add(mrow + 3, v.w, __ATOMIC_RELAXED, __HIP_MEMORY_SCOPE_AGENT);
    if (count_deg && lane == 0)
        __hip_atomic_fetch_add(deg + d, 1.0f, __ATOMIC_RELAXED, __HIP_MEMORY_SCOPE_AGENT);
}

// ---------------------------------------------------------------------------
// Fused SAGE layer GEMM:
//   out[16-node tile] = relu?( x_tile @ Wself + (msg_tile/deg) @ Wneigh + b )
// Block = 128 threads = 4 wave32. Each wave owns one 16-column output tile.
// fp32 WMMA (V_WMMA_F32_16X16X4_F32): preserves reference fp32 numerics; the
// problem is bandwidth-bound so the fp32 matrix path is already far off the
// critical path.
// A fragments come from LDS (row stride FIN+4 dwords -> conflict-free b64
// loads); B fragments are coalesced 16-lane global loads of weight rows.
// ---------------------------------------------------------------------------
template <int FIN, int FOUT, bool RELU>
__global__ __launch_bounds__(128)
void sage_gemm_kernel(const float* __restrict__ hs,     // [N, FIN] self features
                      const float* __restrict__ msg,    // [N, FIN] summed messages
                      const float* __restrict__ deg,    // [N] float degree
                      const float* __restrict__ Wself,  // [FIN, FOUT]
                      const float* __restrict__ Wneigh, // [FIN, FOUT]
                      const float* __restrict__ bias,   // [FOUT]
                      float* __restrict__ out)          // [N, FOUT]
{
    constexpr int LDSS = FIN + 4;                  // pad 4 dwords: kills bank conflicts
    __shared__ float ldsA[2][16][LDSS];            // [0]=self, [1]=mean-neighbor

    const int t = threadIdx.x;
    const long node0 = (long)blockIdx.x * 16;

    // Cooperative stage of the 16xFIN tiles (8 threads/row, float4 chunks),
    // folding the mean (divide by max(deg,1)) into the neighbor tile.
    {
        const int row = t >> 3;
        const int c0 = (t & 7) * (FIN / 8);
        const float* xr = hs + (node0 + row) * FIN + c0;
        const float* mr = msg + (node0 + row) * FIN + c0;
        const float inv = 1.0f / fmaxf(deg[node0 + row], 1.0f);
#pragma unroll
        for (int i = 0; i < FIN / 8; i += 4) {
            const float4 vx = *(const float4*)(xr + i);
            const float4 vm = *(const float4*)(mr + i);
            ldsA[0][row][c0 + i + 0] = vx.x;
            ldsA[0][row][c0 + i + 1] = vx.y;
            ldsA[0][row][c0 + i + 2] = vx.z;
            ldsA[0][row][c0 + i + 3] = vx.w;
            ldsA[1][row][c0 + i + 0] = vm.x * inv;
            ldsA[1][row][c0 + i + 1] = vm.y * inv;
            ldsA[1][row][c0 + i + 2] = vm.z * inv;
            ldsA[1][row][c0 + i + 3] = vm.w * inv;
        }
    }
    __syncthreads();

    const int lane = t & 31;
    const int wv = t >> 5;
    const int m = lane & 15;              // A row / B,D column within tile
    const int kk = (lane >> 4) * 2;       // K sub-offset per lane half (A/B layout)
    const int n_base = (blockIdx.y * 4 + wv) * 16;

    v8f acc = {};                         // 16x16 f32 accumulator (8 VGPRs)

    const float* Ws = Wself + n_base + m;
    const float* Wn = Wneigh + n_base + m;

#pragma unroll 4
    for (int k = 0; k < FIN; k += 4) {
        // A 16x4 f32 fragment: lane holds {A[m][k+kk], A[m][k+kk+1]}
        const v2f aS = *(const v2f*)&ldsA[0][m][k + kk];
        const v2f aN = *(const v2f*)&ldsA[1][m][k + kk];
        // B 4x16 f32 fragment: lane holds {B[k+kk][n], B[k+kk+1][n]}, n = n_base+m
        v2f bS, bN;
        bS.x = Ws[(k + kk + 0) * FOUT];
        bS.y = Ws[(k + kk + 1) * FOUT];
        bN.x = Wn[(k + kk + 0) * FOUT];
        bN.y = Wn[(k + kk + 1) * FOUT];
        acc = __builtin_amdgcn_wmma_f32_16x16x4_f32(false, aS, false, bS,
                                                    (short)0, acc, false, false);
        acc = __builtin_amdgcn_wmma_f32_16x16x4_f32(false, aN, false, bN,
                                                    (short)0, acc, false, false);
    }

    // Epilogue: bias (+ReLU), store. D layout: VGPR r, lanes 0-15 -> row r,
    // lanes 16-31 -> row r+8; column = n_base + (lane&15).
    const float bb = bias[n_base + m];
#pragma unroll
    for (int r = 0; r < 8; r++) {
        const int row = r + ((lane >> 4) << 3);
        float v = acc[r] + bb;
        if (RELU) v = fmaxf(v, 0.0f);
        out[(node0 + row) * FOUT + n_base + m] = v;
    }
}

// ---------------------------------------------------------------------------
// Launch
// ---------------------------------------------------------------------------
extern "C" void kernel_launch(void* const* d_in, const int* in_sizes, int n_in,
                              void* d_out, int out_size, void* d_ws, size_t ws_size,
                              hipStream_t stream) {
    const float* x        = (const float*)d_in[0];
    const int*   edge_idx = (const int*)d_in[1];     // [2, N_EDGES]
    const float* W_self1  = (const float*)d_in[2];
    const float* W_neigh1 = (const float*)d_in[3];
    const float* b1       = (const float*)d_in[4];
    const float* W_self2  = (const float*)d_in[5];
    const float* W_neigh2 = (const float*)d_in[6];
    const float* b2       = (const float*)d_in[7];
    float* out = (float*)d_out;

    const int* src = edge_idx;
    const int* dst = edge_idx + N_EDGES;

    // Workspace layout: msg (reused both layers) | h | deg
    float* msg = (float*)d_ws;
    float* h   = msg + (size_t)N_NODES * IN_F;
    float* deg = h + (size_t)N_NODES * HID_F;

    const int msg4 = N_NODES * IN_F / 4;
    const int deg4 = N_NODES / 4;
    const int scatter_blocks = (N_EDGES + 7) / 8;    // 8 waves (edges) per block
    const int node_tiles = N_NODES / 16;             // 6250, exact

    // ---- Layer 1 ----
    zero_kernel<<<(msg4 + 255) / 256, 256, 0, stream>>>((float4*)msg, msg4);
    zero_kernel<<<(deg4 + 255) / 256, 256, 0, stream>>>((float4*)deg, deg4);
    scatter_kernel<<<scatter_blocks, 256, 0, stream>>>(x, src, dst, msg, deg,
                                                       N_EDGES, 1);
    sage_gemm_kernel<IN_F, HID_F, true>
        <<<dim3(node_tiles, HID_F / 64), 128, 0, stream>>>(
            x, msg, deg, W_self1, W_neigh1, b1, h);

    // ---- Layer 2 (deg reused; same edge list) ----
    zero_kernel<<<(msg4 + 255) / 256, 256, 0, stream>>>((float4*)msg, msg4);
    scatter_kernel<<<scatter_blocks, 256, 0, stream>>>(h, src, dst, msg, deg,
                                                       N_EDGES, 0);
    sage_gemm_kernel<HID_F, OUT_F, false>
        <<<dim3(node_tiles, OUT_F / 64), 128, 0, stream>>>(
            h, msg, deg, W_self2, W_neigh2, b2, out);
}